// RelativePositionMultiHeadAttention_12326556139724
// MI455X (gfx1250) — compile-verified
//
#include <hip/hip_runtime.h>
#include <cstdint>
#include <cstddef>

// ---------------------------------------------------------------------------
// RelativePositionMultiHeadAttention for MI455X (gfx1250), wave32 + WMMA f32.
// B=8, T=1024, M=512, E=64, H=8.
// Matrix math: V_WMMA_F32_16X16X4_F32 (f32-exact vs reference).
// Data movement: GLOBAL_LOAD_ASYNC_TO_LDS_B128 (ASYNCcnt) double-buffered.
// ---------------------------------------------------------------------------

typedef float v2f __attribute__((ext_vector_type(2)));
typedef float v8f __attribute__((ext_vector_type(8)));

constexpr int Bb = 8;      // batch
constexpr int Ts = 1024;   // sequence length
constexpr int Mm = 512;    // model dim
constexpr int Ee = 64;     // head dim
constexpr int Hh = 8;      // heads
constexpr int BT = Bb * Ts;   // 8192 rows
constexpr int EH = Ee * Hh;   // 512 cols
constexpr int PJ = 2 * Ts;    // rel-pos table rows (j = k-q+1023 in [0,2046])

// D = A(16x4) * B(4x16) + C, f32. A/B frags are 2 VGPRs, C/D are 8 VGPRs.
// Per ISA 7.12.2: A: lane l holds A[l%16][2*(l/16)+{0,1}];
//                 B: lane l holds B[2*(l/16)+{0,1}][l%16];
//                 C/D: lane l, vgpr r holds D[r + 8*(l/16)][l%16].
__device__ __forceinline__ v8f wmma4(v2f a, v2f b, v8f c) {
  return __builtin_amdgcn_wmma_f32_16x16x4_f32(
      /*neg_a=*/false, a, /*neg_b=*/false, b,
      /*c_mod=*/(short)0, c, /*reuse_a=*/false, /*reuse_b=*/false);
}

// Async global -> LDS copy, 16 bytes per lane, tracked by ASYNCcnt.
// The low 32 bits of a flat shared-aperture pointer equal the LDS byte
// offset (aperture check uses addr[63:32] only), so truncation is the
// correct VDST operand.
__device__ __forceinline__ void async_ld128(void* lds, const void* gaddr) {
  unsigned l = (unsigned)(size_t)lds;
  unsigned long long g = (unsigned long long)(size_t)gaddr;
  asm volatile("global_load_async_to_lds_b128 %0, %1, off"
               :: "v"(l), "v"(g) : "memory");
}
__device__ __forceinline__ void wait_async0() {
  asm volatile("s_wait_asynccnt 0x0" ::: "memory");
}

// ---------------------------------------------------------------------------
// Kernel 0: relative sinusoid table Prel[j][e], j = rel + (T-1).
// Replaces the 256MB [T,T,E] tensor of the reference with a 512KB table.
// ---------------------------------------------------------------------------
__global__ void prel_kernel(float* __restrict__ P) {
  int idx = blockIdx.x * blockDim.x + threadIdx.x;
  if (idx >= PJ * Ee) return;
  int j = idx / Ee, e = idx % Ee;
  float rel = (float)(j - (Ts - 1));
  float expo = (float)(e & ~1) / (float)Ee;
  float inv = powf(10000.0f, expo);
  float val = rel / inv;
  P[idx] = (e & 1) ? cosf(val) : sinf(val);
}

// ---------------------------------------------------------------------------
// Kernel 1/4: tiled f32 GEMM  C[BT x 512] = A[BT x 512] * B[512 x 512].
// Block = 128 threads (4 waves) -> 32x32 tile. K chunks of 32 staged with
// async-to-LDS, double buffered so the copy of chunk i+1 overlaps the WMMA
// of chunk i. Row stride 36 floats keeps b128 stores 16B-aligned and the
// A-fragment column reads bank-conflict-free (36*r mod 64 has period 16).
// scatter=1 reshapes output column n=e*H+h into [B,H,T,E] (for K/V).
// ---------------------------------------------------------------------------
__global__ __launch_bounds__(128)
void gemm_kernel(const float* __restrict__ A, const float* __restrict__ Bm,
                 float* __restrict__ C, int scatter) {
  __shared__ float As[2][32][36];
  __shared__ float Bs[2][32][36];
  const int tid = threadIdx.x;
  const int wv = tid >> 5, ln = tid & 31;
  const int wm = wv >> 1, wn = wv & 1;          // 2x2 wave tiles
  const int row0 = blockIdx.y * 32;
  const int col0 = blockIdx.x * 32;
  const int n  = ln & 15;
  const int kb = (ln >> 4) << 1;
  const int hi = ln >> 4;

  // Each thread moves 2 float4 of A and 2 of B per chunk (32x32 floats).
  const int r0 = tid >> 3,           c40 = (tid & 7) << 2;
  const int r1 = (tid + 128) >> 3,   c41 = ((tid + 128) & 7) << 2;

  auto stage = [&](int buf, int kc) {
    async_ld128(&As[buf][r0][c40], A + (size_t)(row0 + r0) * Mm + kc + c40);
    async_ld128(&As[buf][r1][c41], A + (size_t)(row0 + r1) * Mm + kc + c41);
    async_ld128(&Bs[buf][r0][c40], Bm + (size_t)(kc + r0) * EH + col0 + c40);
    async_ld128(&Bs[buf][r1][c41], Bm + (size_t)(kc + r1) * EH + col0 + c41);
  };

  stage(0, 0);
  v8f acc = {};
  for (int kc = 0; kc < Mm; kc += 32) {
    const int buf = (kc >> 5) & 1;
    wait_async0();          // own async loads for this chunk done
    __syncthreads();        // -> whole tile visible to all waves
    if (kc + 32 < Mm) stage(buf ^ 1, kc + 32);
#pragma unroll
    for (int s = 0; s < 8; ++s) {
      int kk = s * 4 + kb;
      v2f a; a[0] = As[buf][wm * 16 + n][kk]; a[1] = As[buf][wm * 16 + n][kk + 1];
      v2f b; b[0] = Bs[buf][kk][wn * 16 + n]; b[1] = Bs[buf][kk + 1][wn * 16 + n];
      acc = wmma4(a, b, acc);
    }
  }

#pragma unroll
  for (int r = 0; r < 8; ++r) {
    int mrow = row0 + wm * 16 + r + hi * 8;
    int ncol = col0 + wn * 16 + n;
    float val = acc[r];
    if (scatter) {                              // [BT x EH] -> [B,H,T,E]
      int bI = mrow / Ts, t = mrow % Ts;
      int e = ncol / Hh, h = ncol % Hh;
      C[((size_t)(bI * Hh + h) * Ts + t) * Ee + e] = val;
    } else {
      C[(size_t)mrow * EH + ncol] = val;
    }
  }
}

// ---------------------------------------------------------------------------
// Kernel 2: fused flash-style attention.
// Block = 128 threads (4 waves), each wave owns a 16-row q-tile; K/V 16x64
// tiles async-staged double-buffered in LDS. Per k-tile and wave:
//   S_c  = (Qp+bu) Kt^T                       (16 WMMA)
//   R_t  = (Qp+bv) Prel^T[jw0 .. jw0+32)      (32 WMMA, rel-shift trick)
//   S    = (S_c + gather(R_t)) / sqrt(E), masked, online softmax
//   O   += P * Vt                             (16 WMMA)
// Row stride 68 floats: 272B (16B-aligned b128 stores), and K^T fragment
// reads hit banks 4n+e -> conflict-free.
// ---------------------------------------------------------------------------
__global__ __launch_bounds__(128)
void attn_kernel(const float* __restrict__ Qp,   // [B,T,E*H]
                 const float* __restrict__ Kp,   // [B,H,T,E]
                 const float* __restrict__ Vp,   // [B,H,T,E]
                 const float* __restrict__ Prel, // [2T, E]
                 const float* __restrict__ bu,   // [E,H]
                 const float* __restrict__ bv,   // [E,H]
                 const unsigned char* __restrict__ mask, // [B,T]
                 float* __restrict__ X) {        // [B,T,E,H]
  __shared__ float Kld[2][16][68];
  __shared__ float Vld[2][16][68];
  __shared__ float Rld[4][16][33];
  __shared__ float Sld[4][16][17];
  __shared__ float Ald[4][16];
  __shared__ float Lld[4][16];

  const int tid = threadIdx.x;
  const int wv = tid >> 5, ln = tid & 31;
  const int b = blockIdx.z, h = blockIdx.y;
  const int q0 = blockIdx.x * 64 + wv * 16;
  const int n  = ln & 15;
  const int kb = (ln >> 4) << 1;
  const int hi = ln >> 4;
  const float NEG_INF = -__builtin_inff();

  // Preload Q A-fragments with bu/bv biases folded in (held in VGPRs for the
  // whole k-loop: 64 VGPRs, reused 64 times each).
  v2f quA[16], qvA[16];
  {
    const float* qrow = Qp + ((size_t)b * Ts + q0 + n) * EH + h;
#pragma unroll
    for (int s = 0; s < 16; ++s) {
      int e0 = s * 4 + kb;
      float x0 = qrow[(size_t)e0 * Hh];
      float x1 = qrow[(size_t)(e0 + 1) * Hh];
      quA[s][0] = x0 + bu[e0 * Hh + h];
      quA[s][1] = x1 + bu[(e0 + 1) * Hh + h];
      qvA[s][0] = x0 + bv[e0 * Hh + h];
      qvA[s][1] = x1 + bv[(e0 + 1) * Hh + h];
    }
  }

  const float* Kbh = Kp + (size_t)(b * Hh + h) * Ts * Ee;
  const float* Vbh = Vp + (size_t)(b * Hh + h) * Ts * Ee;

  // Each thread moves 2 float4 of K and 2 of V per 16x64 tile.
  const int sr0 = tid >> 4,          sc0 = (tid & 15) << 2;
  const int sr1 = (tid + 128) >> 4,  sc1 = ((tid + 128) & 15) << 2;
  auto stageKV = [&](int buf, int k0) {
    async_ld128(&Kld[buf][sr0][sc0], Kbh + (size_t)(k0 + sr0) * Ee + sc0);
    async_ld128(&Kld[buf][sr1][sc1], Kbh + (size_t)(k0 + sr1) * Ee + sc1);
    async_ld128(&Vld[buf][sr0][sc0], Vbh + (size_t)(k0 + sr0) * Ee + sc0);
    async_ld128(&Vld[buf][sr1][sc1], Vbh + (size_t)(k0 + sr1) * Ee + sc1);
  };

  v8f acc[4];
#pragma unroll
  for (int ct = 0; ct < 4; ++ct) acc[ct] = {};
  float mrun = NEG_INF, lrun = 0.0f;   // row stats live on lanes 0..15

  stageKV(0, 0);
  for (int k0 = 0; k0 < Ts; k0 += 16) {
    const int buf = (k0 >> 4) & 1;
    wait_async0();          // this wave's async loads for tile k0 done
    __syncthreads();        // all waves' loads visible
    if (k0 + 16 < Ts) stageKV(buf ^ 1, k0 + 16);  // prefetch next tile

    // Content logits: B-frag of K^T -> B[e][k] = Kld[k][e]
    v8f sc = {};
#pragma unroll
    for (int s = 0; s < 16; ++s) {
      int e0 = s * 4 + kb;
      v2f bk; bk[0] = Kld[buf][n][e0]; bk[1] = Kld[buf][n][e0 + 1];
      sc = wmma4(quA[s], bk, sc);
    }

    // Position logits over sliding window jw0 = k0-q0+1008; needed
    // jj = kk + 15 - mq lies in [0,30] of this 32-wide window.
    const int jw0 = k0 - q0 + (Ts - 16);
    v8f rt0 = {}, rt1 = {};
    {
      const float* p0 = Prel + (size_t)(jw0 + n) * Ee;
      const float* p1 = Prel + (size_t)(jw0 + 16 + n) * Ee;
#pragma unroll
      for (int s = 0; s < 16; ++s) {
        int e0 = s * 4 + kb;
        v2f bp; bp[0] = p0[e0]; bp[1] = p0[e0 + 1];
        rt0 = wmma4(qvA[s], bp, rt0);
        v2f bq; bq[0] = p1[e0]; bq[1] = p1[e0 + 1];
        rt1 = wmma4(qvA[s], bq, rt1);
      }
    }
#pragma unroll
    for (int r = 0; r < 8; ++r) {
      int mq = r + hi * 8;
      Rld[wv][mq][n]      = rt0[r];
      Rld[wv][mq][16 + n] = rt1[r];
    }
    __syncthreads();

    // Combine + scale + mask, store S tile for the row-wise softmax.
    const unsigned char mk = mask[(size_t)b * Ts + k0 + n];
#pragma unroll
    for (int r = 0; r < 8; ++r) {
      int mq = r + hi * 8;
      int jj = n + 15 - mq;                       // rel-shift gather
      float sval = (sc[r] + Rld[wv][mq][jj]) * 0.125f;  // / sqrt(E)
      Sld[wv][mq][n] = mk ? sval : NEG_INF;
    }
    __syncthreads();

    // Online softmax row update (one row per lane, lanes 0..15).
    if (ln < 16) {
      int row = ln;
      float mloc = NEG_INF;
#pragma unroll
      for (int kk = 0; kk < 16; ++kk) mloc = fmaxf(mloc, Sld[wv][row][kk]);
      float mnew = fmaxf(mrun, mloc);
      float lsum = 0.0f;
#pragma unroll
      for (int kk = 0; kk < 16; ++kk) {
        float p = __expf(Sld[wv][row][kk] - mnew);
        Sld[wv][row][kk] = p;
        lsum += p;
      }
      float alpha = __expf(mrun - mnew);
      lrun = lrun * alpha + lsum;
      mrun = mnew;
      Ald[wv][row] = alpha;
    }
    __syncthreads();

    // Rescale accumulators and add P * V.
    float al[8];
#pragma unroll
    for (int r = 0; r < 8; ++r) al[r] = Ald[wv][r + hi * 8];
#pragma unroll
    for (int ct = 0; ct < 4; ++ct)
#pragma unroll
      for (int r = 0; r < 8; ++r) acc[ct][r] *= al[r];

#pragma unroll
    for (int s = 0; s < 4; ++s) {
      v2f pa; pa[0] = Sld[wv][n][s * 4 + kb]; pa[1] = Sld[wv][n][s * 4 + kb + 1];
#pragma unroll
      for (int ct = 0; ct < 4; ++ct) {
        v2f bvv;
        bvv[0] = Vld[buf][s * 4 + kb][ct * 16 + n];
        bvv[1] = Vld[buf][s * 4 + kb + 1][ct * 16 + n];
        acc[ct] = wmma4(pa, bvv, acc[ct]);
      }
    }
  }

  // Final normalization and store X in [B,T,E,H] (flat [BT x EH] row-major,
  // ready for the output GEMM against wo flattened as [(e*H+h) x M]).
  if (ln < 16) Lld[wv][ln] = lrun;
  __syncthreads();
  float li[8];
#pragma unroll
  for (int r = 0; r < 8; ++r) li[r] = 1.0f / Lld[wv][r + hi * 8];

#pragma unroll
  for (int ct = 0; ct < 4; ++ct) {
    int e = ct * 16 + n;
#pragma unroll
    for (int r = 0; r < 8; ++r) {
      int qrow = q0 + r + hi * 8;
      X[(((size_t)b * Ts + qrow) * Ee + e) * Hh + h] = acc[ct][r] * li[r];
    }
  }
}

// ---------------------------------------------------------------------------
// Host-side launch. Workspace layout (floats):
//   Qp [BT*EH] | Kp [BT*EH] | Vp [BT*EH] | X [BT*EH] | Prel [2048*64]
// Total ~64.5 MB.
// ---------------------------------------------------------------------------
extern "C" void kernel_launch(void* const* d_in, const int* in_sizes, int n_in,
                              void* d_out, int out_size, void* d_ws, size_t ws_size,
                              hipStream_t stream) {
  (void)in_sizes; (void)n_in; (void)out_size; (void)ws_size;
  const float* q   = (const float*)d_in[0];
  const float* k   = (const float*)d_in[1];
  const float* v   = (const float*)d_in[2];
  const unsigned char* mask = (const unsigned char*)d_in[3];
  const float* wq  = (const float*)d_in[4];
  const float* wk  = (const float*)d_in[5];
  const float* wv  = (const float*)d_in[6];
  const float* bu  = (const float*)d_in[7];
  const float* bv  = (const float*)d_in[8];
  const float* wo  = (const float*)d_in[9];
  float* out = (float*)d_out;

  float* ws   = (float*)d_ws;
  const size_t mat = (size_t)BT * EH;   // 4,194,304 floats
  float* Qp   = ws;
  float* Kp   = Qp + mat;
  float* Vp   = Kp + mat;
  float* X    = Vp + mat;
  float* Prel = X + mat;                // 2048*64 floats

  prel_kernel<<<(PJ * Ee + 255) / 256, 256, 0, stream>>>(Prel);

  dim3 ggrid(EH / 32, BT / 32);         // (16, 256)
  gemm_kernel<<<ggrid, 128, 0, stream>>>(q, wq, Qp, 0);  // [B,T,EH]
  gemm_kernel<<<ggrid, 128, 0, stream>>>(k, wk, Kp, 1);  // [B,H,T,E]
  gemm_kernel<<<ggrid, 128, 0, stream>>>(v, wv, Vp, 1);  // [B,H,T,E]

  dim3 agrid(Ts / 64, Hh, Bb);          // (16, 8, 8)
  attn_kernel<<<agrid, 128, 0, stream>>>(Qp, Kp, Vp, Prel, bu, bv, mask, X);

  gemm_kernel<<<ggrid, 128, 0, stream>>>(X, wo, out, 0); // [B,T,M]
}